// stmffn_57286273794576
// MI455X (gfx1250) — compile-verified
//
#include <hip/hip_runtime.h>

// Problem constants (from the reference)
#define B_   64
#define C_   32
#define V_   1024
#define L_   12
#define O_   32
#define CCAT 224      // (2*3+1)*C
#define KB   32       // K-slab staged in LDS per iteration
#define WT   16       // w-tile width (WMMA M)

// Workspace layout (floats):  [adp: V*V][H1: B*C*V*L]  ~= 105 MB total.
// Everything fits comfortably in the 192 MB L2, so the H1 round-trip is cheap.

typedef __attribute__((ext_vector_type(2))) float v2f;
typedef __attribute__((ext_vector_type(8))) float v8f;

// ---------------------------------------------------------------------------
// adp = softmax(relu(n1 @ n2), axis=1) ;  n1:[V,10], n2:[10,V]
// one block per row r; 256 threads, 4 columns each
// ---------------------------------------------------------------------------
__global__ __launch_bounds__(256) void adp_kernel(const float* __restrict__ n1,
                                                  const float* __restrict__ n2,
                                                  float* __restrict__ adp) {
    __shared__ float red[256];
    __shared__ float rowv[10];
    const int r = blockIdx.x, tid = threadIdx.x;
    if (tid < 10) rowv[tid] = n1[r * 10 + tid];
    __syncthreads();

    float vals[4];
    float mx = -1e30f;
#pragma unroll
    for (int j = 0; j < 4; ++j) {
        const int w = tid * 4 + j;
        float s = 0.f;
#pragma unroll
        for (int k = 0; k < 10; ++k) s += rowv[k] * n2[k * V_ + w];
        s = fmaxf(s, 0.f);          // relu
        vals[j] = s;
        mx = fmaxf(mx, s);
    }
    // block max
    red[tid] = mx; __syncthreads();
    for (int off = 128; off > 0; off >>= 1) {
        if (tid < off) red[tid] = fmaxf(red[tid], red[tid + off]);
        __syncthreads();
    }
    mx = red[0]; __syncthreads();
    // exp + block sum
    float sum = 0.f;
#pragma unroll
    for (int j = 0; j < 4; ++j) { vals[j] = __expf(vals[j] - mx); sum += vals[j]; }
    red[tid] = sum; __syncthreads();
    for (int off = 128; off > 0; off >>= 1) {
        if (tid < off) red[tid] += red[tid + off];
        __syncthreads();
    }
    const float inv = 1.f / red[0];
#pragma unroll
    for (int j = 0; j < 4; ++j) adp[r * V_ + tid * 4 + j] = vals[j] * inv;
}

// ---------------------------------------------------------------------------
// out[b,o,v,l] = bias[o] + sum_c W[o,c] * x[b,c,v,l]   (identity block of h)
// pure overwrite: also clears poison before the accumulating hop kernels
// ---------------------------------------------------------------------------
__global__ __launch_bounds__(256) void init_mix_kernel(const float* __restrict__ x,
                                                       const float* __restrict__ W,
                                                       const float* __restrict__ bias,
                                                       float* __restrict__ out) {
    const int pid = blockIdx.x * 256 + threadIdx.x;
    if (pid >= B_ * V_ * L_) return;
    const int l = pid % L_;
    const int v = (pid / L_) % V_;
    const int b = pid / (V_ * L_);

    float h[C_];
#pragma unroll
    for (int c = 0; c < C_; ++c) h[c] = x[((size_t)(b * C_ + c) * V_ + v) * L_ + l];
#pragma unroll 4
    for (int o = 0; o < O_; ++o) {
        float s = bias[o];
#pragma unroll
        for (int c = 0; c < C_; ++c) s += W[o * CCAT + c] * h[c];
        out[((size_t)(b * O_ + o) * V_ + v) * L_ + l] = s;
    }
}

// ---------------------------------------------------------------------------
// Hop + fused channel-mix kernel.
//   hop[b,c,w,l] = sum_v a[v,w] * xin[b,c,v,l]        (WMMA: M=w, N=l, K=v)
//   out[b,o,w0+w,l] += sum_c W[o, wcolbase+c] * hop[b,c,w,l]
// Block = (w-tile, b); 8 waves x 4 channels cover all 32 channels, so the
// block owns out[b, :, w-tile, :] exclusively -> no atomics needed.
// ---------------------------------------------------------------------------
__global__ __launch_bounds__(256) void hopmix_kernel(const float* __restrict__ xin,
                                                     const float* __restrict__ a,
                                                     const float* __restrict__ W,
                                                     float* __restrict__ out,
                                                     float* __restrict__ hout,
                                                     int wcolbase) {
    union Smem {
        struct {
            float a_slab[KB * WT];        //  2 KB : a[k-slab][w-tile]
            float x_slab[C_ * KB * L_];   // 48 KB : x[b, all c, k-slab, :]
        } s;
        float h[C_ * 16 * 16];            // 32 KB : hop tiles for the mix
    };
    __shared__ Smem sm;

    const int tid   = threadIdx.x;
    const int lane  = tid & 31;
    const int wave  = tid >> 5;          // 0..7
    const int cbase = wave * 4;          // 4 channels per wave
    const int w0    = blockIdx.x * WT;
    const int b     = blockIdx.y;

    const int m   = lane & 15;                    // A row (w) / B col (l)
    const int kv  = (lane < 16) ? 0 : 2;          // K sub-index per ISA layout
    const int lcl = (m < L_) ? m : (L_ - 1);      // clamp l; cols 12..15 unused

    v8f acc0 = {}, acc1 = {}, acc2 = {}, acc3 = {};

    for (int k0 = 0; k0 < V_; k0 += KB) {
        __syncthreads();
        // stage a[k0:k0+KB, w0:w0+16] -> LDS (float4, 128 threads)
        if (tid < (KB * WT / 4)) {
            const int r = tid >> 2, q = tid & 3;
            *(float4*)&sm.s.a_slab[r * WT + q * 4] =
                *(const float4*)&a[(size_t)(k0 + r) * V_ + w0 + q * 4];
        }
        // stage xin[b, 0:32, k0:k0+KB, 0:12] -> LDS (3x float4 per row)
        for (int rr = tid; rr < C_ * KB; rr += 256) {
            const int c = rr >> 5, r = rr & (KB - 1);
            const float4* src =
                (const float4*)&xin[((size_t)(b * C_ + c) * V_ + (k0 + r)) * L_];
            float4* dst = (float4*)&sm.s.x_slab[(c * KB + r) * L_];
            dst[0] = src[0]; dst[1] = src[1]; dst[2] = src[2];
        }
        if (k0 + KB < V_)  // hint the next a-slab into cache
            __builtin_prefetch(&a[(size_t)(k0 + KB) * V_ + w0], 0, 0);
        __syncthreads();

#pragma unroll
        for (int kk = 0; kk < KB; kk += 4) {
            // A operand: Amat[m][k] = a[v, w0+m]   (16x4 f32, 2 VGPRs/lane)
            v2f A;
            A.x = sm.s.a_slab[(kk + kv + 0) * WT + m];
            A.y = sm.s.a_slab[(kk + kv + 1) * WT + m];
#pragma unroll
            for (int j = 0; j < 4; ++j) {
                const int c = cbase + j;
                v2f Bm;
                Bm.x = sm.s.x_slab[(c * KB + kk + kv + 0) * L_ + lcl];
                Bm.y = sm.s.x_slab[(c * KB + kk + kv + 1) * L_ + lcl];
                if (j == 0) acc0 = __builtin_amdgcn_wmma_f32_16x16x4_f32(false, A, false, Bm, (short)0, acc0, false, false);
                if (j == 1) acc1 = __builtin_amdgcn_wmma_f32_16x16x4_f32(false, A, false, Bm, (short)0, acc1, false, false);
                if (j == 2) acc2 = __builtin_amdgcn_wmma_f32_16x16x4_f32(false, A, false, Bm, (short)0, acc2, false, false);
                if (j == 3) acc3 = __builtin_amdgcn_wmma_f32_16x16x4_f32(false, A, false, Bm, (short)0, acc3, false, false);
            }
        }
    }

    __syncthreads();
    // D layout: VGPR i, lanes 0-15 -> (w=i, l=lane); lanes 16-31 -> (w=8+i)
    {
        const int n = lane & 15;
        const int wrow = (lane < 16) ? 0 : 8;
#pragma unroll
        for (int i = 0; i < 8; ++i) {
            sm.h[(cbase + 0) * 256 + (wrow + i) * 16 + n] = acc0[i];
            sm.h[(cbase + 1) * 256 + (wrow + i) * 16 + n] = acc1[i];
            sm.h[(cbase + 2) * 256 + (wrow + i) * 16 + n] = acc2[i];
            sm.h[(cbase + 3) * 256 + (wrow + i) * 16 + n] = acc3[i];
        }
        // materialize hop-1 output (input of the next hop); hop-2 skips this
        if (hout != nullptr && n < L_) {
#pragma unroll
            for (int i = 0; i < 8; ++i) {
                const size_t wrv = (size_t)w0 + wrow + i;
                hout[((size_t)(b * C_ + cbase + 0) * V_ + wrv) * L_ + n] = acc0[i];
                hout[((size_t)(b * C_ + cbase + 1) * V_ + wrv) * L_ + n] = acc1[i];
                hout[((size_t)(b * C_ + cbase + 2) * V_ + wrv) * L_ + n] = acc2[i];
                hout[((size_t)(b * C_ + cbase + 3) * V_ + wrv) * L_ + n] = acc3[i];
            }
        }
    }
    __syncthreads();

    // fused 1x1-conv contribution of this channel block into out
    for (int idx = tid; idx < O_ * WT * L_; idx += 256) {
        const int o = idx / (WT * L_);
        const int rem = idx % (WT * L_);
        const int w = rem / L_, l = rem % L_;
        float s = 0.f;
#pragma unroll
        for (int c = 0; c < C_; ++c)
            s += W[o * CCAT + wcolbase + c] * sm.h[c * 256 + w * 16 + l];
        out[((size_t)(b * O_ + o) * V_ + w0 + w) * L_ + l] += s;
    }
}

// ---------------------------------------------------------------------------
extern "C" void kernel_launch(void* const* d_in, const int* in_sizes, int n_in,
                              void* d_out, int out_size, void* d_ws, size_t ws_size,
                              hipStream_t stream) {
    const float* x    = (const float*)d_in[0];
    const float* A1   = (const float*)d_in[1];
    const float* A2   = (const float*)d_in[2];
    const float* n1   = (const float*)d_in[3];
    const float* n2   = (const float*)d_in[4];
    const float* W    = (const float*)d_in[5];
    const float* bmlp = (const float*)d_in[6];
    float* out = (float*)d_out;

    float* adp = (float*)d_ws;                       // V*V        (4 MB)
    float* H1  = adp + (size_t)V_ * V_;              // B*C*V*L  (100 MB)

    adp_kernel<<<V_, 256, 0, stream>>>(n1, n2, adp);
    init_mix_kernel<<<(B_ * V_ * L_ + 255) / 256, 256, 0, stream>>>(x, W, bmlp, out);

    const dim3 grid(V_ / WT, B_);
    const float* sup[3] = {A1, A2, adp};
    for (int s = 0; s < 3; ++s) {
        // hop 1: materialize H1 and fold mix block (1+2s)
        hopmix_kernel<<<grid, 256, 0, stream>>>(x, sup[s], W, out, H1, (1 + 2 * s) * C_);
        // hop 2: consume H1, fold mix block (2+2s), no materialization
        hopmix_kernel<<<grid, 256, 0, stream>>>(H1, sup[s], W, out, nullptr, (2 + 2 * s) * C_);
    }
}